// KD_Encoding_11665131176024
// MI455X (gfx1250) — compile-verified
//
#include <hip/hip_runtime.h>

// KD / product-quantization encoding for MI455X (gfx1250, wave32).
// Forward of the straight-through estimator is exactly y_hard, so this is a
// nearest-centroid search: label = argmin_k (|c|^2 - 2 x.c), then gather the
// winning centroid. Score GEMM uses exact-precision v_wmma_f32_16x16x4_f32.
// Memory floor ~100MB @ 23.3 TB/s (~4.3us). All global accesses are expressed
// as scalar-base + loop-invariant 32-bit lane offset + immediate so they lower
// to GVS addressing (no per-tile 64-bit VALU address math).

typedef float v2f __attribute__((ext_vector_type(2)));
typedef float v4f __attribute__((ext_vector_type(4)));
typedef float v8f __attribute__((ext_vector_type(8)));

#define B_TOT 65536
#define D_TOT 128
#define M_SUB 8
#define K_CNT 256
#define d_SUB 16

// ---------------------------------------------------------------------------
// Prologue: |c|^2 for all 8*256 centroids -> d_ws (8KB, L2-resident).
// ---------------------------------------------------------------------------
__global__ __launch_bounds__(256) void csq_kernel(
    const float* __restrict__ center, float* __restrict__ csq)
{
    int idx = blockIdx.x * 256 + threadIdx.x;          // 0..2047
    int p = idx * d_SUB;
    v4f x0 = *(const v4f*)(center + p);
    v4f x1 = *(const v4f*)(center + p + 4);
    v4f x2 = *(const v4f*)(center + p + 8);
    v4f x3 = *(const v4f*)(center + p + 12);
    float s = 0.f;
#pragma unroll
    for (int j = 0; j < 4; ++j)
        s += x0[j]*x0[j] + x1[j]*x1[j] + x2[j]*x2[j] + x3[j]*x3[j];
    csq[idx] = s;
}

// ---------------------------------------------------------------------------
// Main kernel. Block = 256 threads = 8 wave32s; block owns a 16-row batch
// tile; wave w owns subspace m = w (all 256 centroids, d = 16).
// ---------------------------------------------------------------------------
template <bool USE_CSQ>
__global__ __launch_bounds__(256) void kd_encode_kernel(
    const float* __restrict__ X,       // [B, 128]
    const float* __restrict__ center,  // [8, 256, 16]
    const float* __restrict__ csq,     // [8, 256] or nullptr
    float* __restrict__ out)           // X_p | X_p_m | center | codes | weight
{
    __shared__ float sval[M_SUB][16 * 17];
    __shared__ int   sidx[M_SUB][16 * 17];

    const int tid  = threadIdx.x;
    const int w    = tid >> 5;   // wave id == subspace m
    const int lane = tid & 31;
    const int lr   = lane & 15;  // A-row (batch row) == B-col (centroid) == D-col
    const int h    = lane >> 4;  // half-wave selector
    const int b0   = blockIdx.x << 4;

    // Loop-invariant 32-bit lane offsets (element indices, all < 2^25):
    const int xoff = (b0 + lr) * D_TOT + w * d_SUB + 2 * h;  // into X
    const int coff = (w * K_CNT + lr) * d_SUB + 2 * h;       // into center
    const int qoff = w * K_CNT + lr;                         // into csq

    // ---- A tile: 16 rows x 16 cols of this subspace, pre-scaled by -2 ----
    // f32 16x4 A layout: lanes 0-15 hold K=(0,1), lanes 16-31 hold K=(2,3)
    // -> per k-step: float2 at column 4*kk + 2*h. X is single-use: NT loads.
    v2f a[4];
#pragma unroll
    for (int kk = 0; kk < 4; ++kk) {
        v2f t = __builtin_nontemporal_load((const v2f*)(X + xoff + 4 * kk));
        a[kk] = t * -2.0f;
    }

    // Running per-component argmin in the native D layout:
    // component i of this lane is (row = i + 8*h, col = lr); candidate index
    // for tile nt is nt*16 + lr, so storing just nt (an inline constant in the
    // unrolled loop) is enough.
    float minv[8];
    int   mint[8];

#pragma unroll
    for (int nt = 0; nt < 16; ++nt) {
        // B[k,n] = center[w, nt*16 + n, 4*kk + k]; same half-wave striping as A.
        v8f   acc = {};
        float c;
        float ssq = 0.0f;
        if (USE_CSQ)
            c = csq[qoff + nt * 16];             // SADDR + imm, L2-resident
#pragma unroll
        for (int kk = 0; kk < 4; ++kk) {
            v2f bv = *(const v2f*)(center + coff + nt * 256 + kk * 4);
            if (!USE_CSQ)
                ssq = fmaf(bv.x, bv.x, fmaf(bv.y, bv.y, ssq));
            acc = __builtin_amdgcn_wmma_f32_16x16x4_f32(
                false, a[kk], false, bv, (short)0, acc, false, false);
        }
        if (!USE_CSQ) {
            ssq += __shfl_xor(ssq, 16);          // partner holds other 8 dims
            c = ssq;
        }
#pragma unroll
        for (int i = 0; i < 8; ++i) {
            float s = acc[i] + c;                // |c|^2 - 2 x.c
            if (nt == 0) {                       // peeled: no compares vs +INF
                minv[i] = s;
                mint[i] = 0;
            } else {
                bool better = s < minv[i];       // strict <: first-index ties
                minv[i] = better ? s  : minv[i];
                mint[i] = better ? nt : mint[i]; // cndmask with inline nt
            }
        }
    }

    // ---- one LDS transpose: (row, col) grid of per-column winners ----
#pragma unroll
    for (int i = 0; i < 8; ++i) {
        sval[w][(i + 8 * h) * 17 + lr] = minv[i];
        sidx[w][(i + 8 * h) * 17 + lr] = mint[i];
    }
    __syncthreads();

    // Both half-wave lanes scan the 16 columns of batch row lr (stride 17:
    // conflict-free) with exact lexicographic (value, index) tie-breaking.
    float bestv = 3.4028235e38f;
    int   besti = 0;
#pragma unroll
    for (int c2 = 0; c2 < 16; ++c2) {
        float v  = sval[w][lr * 17 + c2];
        int   gi = (sidx[w][lr * 17 + c2] << 4) | c2;
        if (v < bestv || (v == bestv && gi < besti)) { bestv = v; besti = gi; }
    }

    // ---- gather winning centroid, write X_p and X_p_m (identical layout) ----
    const int goff = (w * K_CNT + besti) * d_SUB + 8 * h;
    v4f c0 = *(const v4f*)(center + goff);
    v4f c1 = *(const v4f*)(center + goff + 4);

    const int XPM = B_TOT * D_TOT;                       // 8388608 elements
    const int o   = (b0 + lr) * (M_SUB * d_SUB) + w * d_SUB + 8 * h;
    __builtin_nontemporal_store(c0, (v4f*)(out + o));
    __builtin_nontemporal_store(c1, (v4f*)(out + o + 4));
    __builtin_nontemporal_store(c0, (v4f*)(out + XPM + o));
    __builtin_nontemporal_store(c1, (v4f*)(out + XPM + o + 4));

    if (h == 0) {
        const int CODES = 2 * XPM + M_SUB * K_CNT * d_SUB;  // 16809984
        __builtin_nontemporal_store((float)besti,
                                    out + CODES + (b0 + lr) * M_SUB + w);
    }
}

extern "C" void kernel_launch(void* const* d_in, const int* in_sizes, int n_in,
                              void* d_out, int out_size, void* d_ws, size_t ws_size,
                              hipStream_t stream) {
    const float* X      = (const float*)d_in[0];
    const float* center = (const float*)d_in[1];
    const float* weight = (const float*)d_in[2];
    float* out = (float*)d_out;

    const size_t XPM      = (size_t)B_TOT * D_TOT;            // 8388608
    const size_t CENTER_N = (size_t)M_SUB * K_CNT * d_SUB;    // 32768
    const size_t CODES_N  = (size_t)B_TOT * M_SUB;            // 524288
    const size_t WEIGHT_N = (size_t)D_TOT * D_TOT;            // 16384

    // passthrough outputs: center, weight
    hipMemcpyAsync(out + 2 * XPM, center, CENTER_N * sizeof(float),
                   hipMemcpyDeviceToDevice, stream);
    hipMemcpyAsync(out + 2 * XPM + CENTER_N + CODES_N, weight,
                   WEIGHT_N * sizeof(float), hipMemcpyDeviceToDevice, stream);

    const size_t CSQ_BYTES = (size_t)M_SUB * K_CNT * sizeof(float); // 8KB
    if (ws_size >= CSQ_BYTES) {
        float* csq = (float*)d_ws;
        csq_kernel<<<M_SUB * K_CNT / 256, 256, 0, stream>>>(center, csq);
        kd_encode_kernel<true><<<B_TOT / 16, 256, 0, stream>>>(X, center, csq, out);
    } else {
        kd_encode_kernel<false><<<B_TOT / 16, 256, 0, stream>>>(X, center, nullptr, out);
    }
    (void)in_sizes; (void)n_in; (void)out_size;
}